// ConsistentStrongFormLoss_57019985822568
// MI455X (gfx1250) — compile-verified
//
#include <hip/hip_runtime.h>
#include <hip/hip_bf16.h>

// ---------------------------------------------------------------------------
// ConsistentStrongFormLoss for MI455X (gfx1250, wave32).
// Bandwidth-bound graph scatter/gather; no dense GEMM. WMMA f32 16x16x4 is
// used as a one-shot wave reduction engine; prefetch hides gather latency.
// ---------------------------------------------------------------------------

#define W_PDE 1.0
#define W_BC  100.0
#define W_J1  10.0
#define W_J2  10.0

typedef float v2f __attribute__((ext_vector_type(2)));
typedef float v8f __attribute__((ext_vector_type(8)));

// Wave32 sum reduction. On gfx1250, V_WMMA_F32_16X16X4_F32 with
//   A[m][k] : lane m (VGPR0 -> k=0 half 0, k=2 half 1) = per-lane partial
//   B       : all ones (4x16)
// yields D[m][n] = x[m] + x[m+16] for every n. Summing the 8 D VGPRs gives
// half-wave h the sum of lanes 8h..8h+7 and 8h+16..8h+23; one xor-16 shuffle finishes.
__device__ __forceinline__ float wave_reduce_sum(float v) {
#if defined(__gfx1250__) && __has_builtin(__builtin_amdgcn_wmma_f32_16x16x4_f32)
  v2f A; A[0] = v;    A[1] = 0.0f;   // only k = {0,2} slots carry data
  v2f B; B[0] = 1.0f; B[1] = 1.0f;   // ones matrix
  v8f C = {};
  v8f D = __builtin_amdgcn_wmma_f32_16x16x4_f32(
      /*neg_a=*/false, A, /*neg_b=*/false, B,
      /*c_mod=*/(short)0, C, /*reuse_a=*/false, /*reuse_b=*/false);
  float s = (D[0] + D[1]) + (D[2] + D[3]) + (D[4] + D[5]) + (D[6] + D[7]);
  s += __shfl_xor(s, 16, 32);
  return s;
#else
  for (int off = 16; off > 0; off >>= 1) v += __shfl_xor(v, off, 32);
  return v;
#endif
}

// Block-wide sum; result valid on wave 0 (all lanes). 256-thread blocks.
__device__ __forceinline__ float block_reduce_sum(float v, float* smem8) {
  const int lane = threadIdx.x & 31;
  const int wave = threadIdx.x >> 5;
  float s = wave_reduce_sum(v);
  if (lane == 0) smem8[wave] = s;
  __syncthreads();
  float r = 0.0f;
  if (wave == 0) {
    r = (lane < (int)(blockDim.x >> 5)) ? smem8[lane] : 0.0f;
    r = wave_reduce_sum(r);
  }
  __syncthreads();
  return r;
}

__device__ __forceinline__ float sgnf(float z) {
  return (z > 0.0f) ? 1.0f : ((z < 0.0f) ? -1.0f : 0.0f);
}

// acc layout (doubles): 0=sum res_pde^2, 1=sum bc sq, 2=cnt bnd,
//                       3=sum j1 sq, 4=cnt het, 5=sum j2 sq
// ---------------------------------------------------------------------------
// Pass 1 over edges: scatter-add dx/dy/lap, fused loss_j1 reduction.
// ---------------------------------------------------------------------------
__global__ __launch_bounds__(256)
void edge_pass1(const float* __restrict__ u, const float* __restrict__ x,
                const float* __restrict__ j1,
                const float* __restrict__ cdx, const float* __restrict__ cdy,
                const float* __restrict__ clap,
                const int* __restrict__ eidx, const int* __restrict__ eattr,
                float* __restrict__ dx, float* __restrict__ dy,
                float* __restrict__ lap, double* __restrict__ acc, int E) {
  __shared__ float sred[8];
  float j1s = 0.0f, cnt = 0.0f;
  const int stride = gridDim.x * blockDim.x;
  for (int i = blockIdx.x * blockDim.x + threadIdx.x; i < E; i += stride) {
    const int r = eidx[i];
    const int c = eidx[E + i];
    // Prefetch next iteration's random gathers into cache ahead of use.
    const int ip = i + stride;
    if (ip < E) {
      __builtin_prefetch(&u[eidx[ip]], 0, 0);
      __builtin_prefetch(&u[eidx[E + ip]], 0, 0);
    }
    const float du = u[c] - u[r];
    unsafeAtomicAdd(&dx[r],  du * cdx[i]);
    unsafeAtomicAdd(&dy[r],  du * cdy[i]);
    unsafeAtomicAdd(&lap[r], du * clap[i]);
    if (eattr[i] == 1) {
      const float sg = sgnf(x[4 * c + 3]);     // sign(z[dst])
      const float d  = sg * du - j1[r];        // val_jump_pred - j1[src]
      j1s += d * d;
      cnt += 1.0f;
    }
  }
  float bs = block_reduce_sum(j1s, sred);
  if (threadIdx.x == 0) unsafeAtomicAdd(&acc[3], (double)bs);
  bs = block_reduce_sum(cnt, sred);
  if (threadIdx.x == 0) unsafeAtomicAdd(&acc[4], (double)bs);
}

// ---------------------------------------------------------------------------
// Node pass: PDE residual + boundary loss.
// ---------------------------------------------------------------------------
__global__ __launch_bounds__(256)
void node_pass(const float* __restrict__ u, const float* __restrict__ x,
               const float* __restrict__ pos, const float* __restrict__ y,
               const float* __restrict__ src, const float* __restrict__ lap,
               const float* __restrict__ bm_p, const float* __restrict__ bp_p,
               double* __restrict__ acc, int N) {
  __shared__ float sred[8];
  const float bm = bm_p[0], bp = bp_p[0];
  float pde = 0.0f, bc = 0.0f, bcc = 0.0f;
  const int stride = gridDim.x * blockDim.x;
  for (int n = blockIdx.x * blockDim.x + threadIdx.x; n < N; n += stride) {
    const float z    = x[4 * n + 3];
    const float beta = (z < 0.0f) ? bm : bp;
    const float res  = -lap[n] - src[n] / beta;
    pde += res * res;
    const float p0 = pos[2 * n], p1 = pos[2 * n + 1];
    if (fabsf(p0) > 0.99f || fabsf(p1) > 0.99f) {
      const float d = u[n] - y[n];
      bc  += d * d;
      bcc += 1.0f;
    }
  }
  float bs = block_reduce_sum(pde, sred);
  if (threadIdx.x == 0) unsafeAtomicAdd(&acc[0], (double)bs);
  bs = block_reduce_sum(bc, sred);
  if (threadIdx.x == 0) unsafeAtomicAdd(&acc[1], (double)bs);
  bs = block_reduce_sum(bcc, sred);
  if (threadIdx.x == 0) unsafeAtomicAdd(&acc[2], (double)bs);
}

// ---------------------------------------------------------------------------
// Pass 2 over edges: flux-jump loss (needs completed dx/dy).
// ---------------------------------------------------------------------------
__global__ __launch_bounds__(256)
void edge_pass2(const float* __restrict__ x, const float* __restrict__ pos,
                const int* __restrict__ eidx, const int* __restrict__ eattr,
                const float* __restrict__ dx, const float* __restrict__ dy,
                const float* __restrict__ a_p, const float* __restrict__ b_p,
                const float* __restrict__ c_p, const float* __restrict__ bm_p,
                const float* __restrict__ bp_p, double* __restrict__ acc,
                int E) {
  __shared__ float sred[8];
  const float av = a_p[0], bv = b_p[0], cv = c_p[0];
  const float bm = bm_p[0], bp = bp_p[0];
  const float inva2 = 1.0f / (av * av);
  const float invb2 = 1.0f / (bv * bv);
  const float tcoef = 2.0f * (cv * bp - bm);
  float j2 = 0.0f;
  const int stride = gridDim.x * blockDim.x;
  for (int i = blockIdx.x * blockDim.x + threadIdx.x; i < E; i += stride) {
    if (eattr[i] != 1) continue;
    const int r = eidx[i];
    const int c = eidx[E + i];
    __builtin_prefetch(&dx[r], 0, 0);
    __builtin_prefetch(&dx[c], 0, 0);
    const float zs = x[4 * r + 3];
    const float zd = x[4 * c + 3];
    const float sg = sgnf(zd);
    const float betas = (zs < 0.0f) ? bm : bp;
    const float betad = (zd < 0.0f) ? bm : bp;
    const float pmx = 0.5f * (pos[2 * r]     + pos[2 * c]);
    const float pmy = 0.5f * (pos[2 * r + 1] + pos[2 * c + 1]);
    const float nx = pmx * inva2;
    const float ny = pmy * invb2;
    const float nrm = fmaxf(sqrtf(nx * nx + ny * ny), 1e-8f);
    const float nxu = nx / nrm, nyu = ny / nrm;
    const float dns = dx[r] * nxu + dy[r] * nyu;
    const float dnd = dx[c] * nxu + dy[c] * nyu;
    const float fj = sg * (betad * dnd - betas * dns);
    const float d = fj - tcoef * nrm;
    j2 += d * d;
  }
  float bs = block_reduce_sum(j2, sred);
  if (threadIdx.x == 0) unsafeAtomicAdd(&acc[5], (double)bs);
}

// ---------------------------------------------------------------------------
// Finalize: combine weighted losses into scalar output.
// ---------------------------------------------------------------------------
__global__ void finalize_kernel(const double* __restrict__ acc,
                                float* __restrict__ out, int N) {
  if (threadIdx.x == 0 && blockIdx.x == 0) {
    const double loss_pde = acc[0] / (double)N;
    const double loss_bc  = acc[1] / fmax(acc[2], 1.0);
    const double loss_j1  = acc[3] / fmax(acc[4], 1.0);
    const double loss_j2  = acc[5] / fmax(acc[4], 1.0);
    out[0] = (float)(W_PDE * loss_pde + W_BC * loss_bc +
                     W_J1 * loss_j1 + W_J2 * loss_j2);
  }
}

extern "C" void kernel_launch(void* const* d_in, const int* in_sizes, int n_in,
                              void* d_out, int out_size, void* d_ws,
                              size_t ws_size, hipStream_t stream) {
  const float* u    = (const float*)d_in[0];
  const float* x    = (const float*)d_in[1];
  const float* pos  = (const float*)d_in[2];
  const float* y    = (const float*)d_in[3];
  const float* src  = (const float*)d_in[4];
  const float* j1   = (const float*)d_in[5];
  const float* cdx  = (const float*)d_in[6];
  const float* cdy  = (const float*)d_in[7];
  const float* clap = (const float*)d_in[8];
  const float* a_p  = (const float*)d_in[9];
  const float* b_p  = (const float*)d_in[10];
  const float* c_p  = (const float*)d_in[11];
  const float* bm_p = (const float*)d_in[12];
  const float* bp_p = (const float*)d_in[13];
  const int* eidx   = (const int*)d_in[14];
  const int* eattr  = (const int*)d_in[15];

  const int N = in_sizes[0];   // u_pred element count
  const int E = in_sizes[6];   // coeff_dx element count

  // Workspace: dx|dy|lap (N floats each) then 6 doubles of accumulators.
  float* dx  = (float*)d_ws;
  float* dy  = dx + N;
  float* lap = dy + N;
  size_t accOff = ((size_t)3 * (size_t)N * sizeof(float) + 15) & ~(size_t)15;
  double* acc = (double*)((char*)d_ws + accOff);
  const size_t needed = accOff + 8 * sizeof(double);
  if (ws_size < needed) return;

  hipMemsetAsync(d_ws, 0, needed, stream);

  const int threads = 256;
  int eBlocks = (E + threads - 1) / threads;
  if (eBlocks > 8192) eBlocks = 8192;
  int nBlocks = (N + threads - 1) / threads;
  if (nBlocks > 8192) nBlocks = 8192;

  edge_pass1<<<eBlocks, threads, 0, stream>>>(u, x, j1, cdx, cdy, clap, eidx,
                                              eattr, dx, dy, lap, acc, E);
  node_pass<<<nBlocks, threads, 0, stream>>>(u, x, pos, y, src, lap, bm_p,
                                             bp_p, acc, N);
  edge_pass2<<<eBlocks, threads, 0, stream>>>(x, pos, eidx, eattr, dx, dy,
                                              a_p, b_p, c_p, bm_p, bp_p, acc,
                                              E);
  finalize_kernel<<<1, 32, 0, stream>>>(acc, (float*)d_out, N);
}